// EmbeddingGemmaPooler_14671608283271
// MI455X (gfx1250) — compile-verified
//
#include <hip/hip_runtime.h>

#define NUM_SEQS 128
#define HIDDEN   768
#define FF       3072
#define SPLITS   8
#define EPS      1e-12f

typedef __attribute__((ext_vector_type(2))) float v2f;
typedef __attribute__((ext_vector_type(4))) float v4f;
typedef __attribute__((ext_vector_type(8))) float v8f;

// ---------------------------------------------------------------------------
// Kernel 1: partial ragged segment-sum.  grid = (NUM_SEQS, SPLITS), block=192.
// Each thread owns 4 consecutive columns (float4); one block iteration reads a
// full 768-float row, coalesced b128.  Non-temporal: hidden_states is 201 MB,
// streamed once — keep it out of L2 so W1/W2 stay resident.
// ---------------------------------------------------------------------------
__global__ void pool_partial_kernel(const float* __restrict__ hs,
                                    const int* __restrict__ lens,
                                    float* __restrict__ partials) {
    const int seg   = blockIdx.x;
    const int split = blockIdx.y;
    // Scalar prefix sum over <=128 ints (uniform -> SALU loop, negligible).
    int start = 0;
    for (int i = 0; i < seg; ++i) start += lens[i];
    const int len   = lens[seg];
    const int chunk = (len + SPLITS - 1) / SPLITS;
    const int r0    = split * chunk;
    const int r1    = (r0 + chunk < len) ? (r0 + chunk) : len;

    const int t = threadIdx.x;                 // 0..191, col = 4*t
    v4f acc = {0.f, 0.f, 0.f, 0.f};
    const v4f* __restrict__ base =
        (const v4f*)hs + (size_t)start * (HIDDEN / 4) + t;
    for (int r = r0; r < r1; ++r) {
        v4f v = __builtin_nontemporal_load(base + (size_t)r * (HIDDEN / 4));
        acc.x += v.x; acc.y += v.y; acc.z += v.z; acc.w += v.w;
    }
    v4f* out = (v4f*)partials + ((size_t)seg * SPLITS + split) * (HIDDEN / 4) + t;
    *out = acc;
}

// ---------------------------------------------------------------------------
// Kernel 2: reduce the SPLITS partials and divide by length -> pooled [128,768]
// ---------------------------------------------------------------------------
__global__ void pool_reduce_kernel(const float* __restrict__ partials,
                                   const int* __restrict__ lens,
                                   float* __restrict__ pooled) {
    const int seg = blockIdx.x;
    const int t   = threadIdx.x;               // 0..191
    v4f acc = {0.f, 0.f, 0.f, 0.f};
    for (int s = 0; s < SPLITS; ++s) {
        v4f v = ((const v4f*)partials)[((size_t)seg * SPLITS + s) * (HIDDEN / 4) + t];
        acc.x += v.x; acc.y += v.y; acc.z += v.z; acc.w += v.w;
    }
    const float inv = 1.0f / (float)lens[seg];
    acc.x *= inv; acc.y *= inv; acc.z *= inv; acc.w *= inv;
    ((v4f*)pooled)[(size_t)seg * (HIDDEN / 4) + t] = acc;
}

// ---------------------------------------------------------------------------
// Kernel 3: fp32 WMMA GEMM, D[M,N] = A[M,K] * W[N,K]^T  (both row-major).
// One wave per 16x16 tile; K loop in steps of 4 via V_WMMA_F32_16X16X4_F32.
// ISA layouts (wave32):
//   A 16x4 : lane L -> M = L&15 ; VGPRs hold K = {0,1} (L<16) / {2,3} (L>=16)
//   B 4x16 : lane L -> N = L&15 ; same K split (symmetric)
//   D 16x16: VGPR r, lane L -> row r + 8*(L>>4), col L&15
// All branching is wave-uniform so EXEC stays all-ones for WMMA.
// ---------------------------------------------------------------------------
__global__ void wmma_gemm_nt_kernel(const float* __restrict__ A,
                                    const float* __restrict__ W,
                                    float* __restrict__ D,
                                    int K, int N) {
    const int lane  = threadIdx.x & 31;
    const int wave  = threadIdx.x >> 5;
    const int tile  = blockIdx.x * (blockDim.x >> 5) + wave;
    const int ntn   = N >> 4;
    const int mt    = tile / ntn;
    const int nt    = tile % ntn;
    if (mt >= (NUM_SEQS >> 4)) return;         // uniform per wave

    const int mn    = lane & 15;
    const int kbase = (lane >> 4) << 1;        // 0 or 2
    const float* __restrict__ arow = A + (size_t)((mt << 4) + mn) * K;
    const float* __restrict__ wrow = W + (size_t)((nt << 4) + mn) * K;

    v8f c = {};
#pragma unroll 4
    for (int k = 0; k < K; k += 4) {
        v2f a = *(const v2f*)(arow + k + kbase);   // 8B aligned: K,k,kbase even
        v2f b = *(const v2f*)(wrow + k + kbase);
        c = __builtin_amdgcn_wmma_f32_16x16x4_f32(
                /*neg_a=*/false, a, /*neg_b=*/false, b,
                /*c_mod=*/(short)0, c, /*reuse_a=*/false, /*reuse_b=*/false);
    }
    const int half = (lane >> 4) << 3;          // 0 or 8
#pragma unroll
    for (int r = 0; r < 8; ++r) {
        const int row = (mt << 4) + half + r;
        D[(size_t)row * N + (nt << 4) + mn] = c[r];
    }
}

// ---------------------------------------------------------------------------
// Kernel 4: row-wise L2 normalize (wave32 shfl reduction), [128,768] -> out.
// ---------------------------------------------------------------------------
__global__ void l2norm_rows_kernel(const float* __restrict__ h,
                                   float* __restrict__ out) {
    __shared__ float sred[8];
    __shared__ float sinv;
    const int seg = blockIdx.x;
    const int t   = threadIdx.x;               // 256 threads = 8 waves
    const float* __restrict__ row = h + (size_t)seg * HIDDEN;

    float ss = 0.f;
    for (int c = t; c < HIDDEN; c += 256) {
        float v = row[c];
        ss = fmaf(v, v, ss);
    }
#pragma unroll
    for (int off = 16; off > 0; off >>= 1)
        ss += __shfl_xor(ss, off, 32);
    if ((t & 31) == 0) sred[t >> 5] = ss;
    __syncthreads();
    if (t == 0) {
        float tot = 0.f;
#pragma unroll
        for (int i = 0; i < 8; ++i) tot += sred[i];
        float nrm = sqrtf(tot);
        sinv = 1.0f / fmaxf(nrm, EPS);
    }
    __syncthreads();
    const float inv = sinv;
    for (int c = t; c < HIDDEN; c += 256)
        out[(size_t)seg * HIDDEN + c] = row[c] * inv;
}

// ---------------------------------------------------------------------------
extern "C" void kernel_launch(void* const* d_in, const int* in_sizes, int n_in,
                              void* d_out, int out_size, void* d_ws, size_t ws_size,
                              hipStream_t stream) {
    const float* hs   = (const float*)d_in[0];   // [65536, 768] f32
    const int*   lens = (const int*)d_in[1];     // [128] i32
    const float* W1   = (const float*)d_in[2];   // [3072, 768] f32
    const float* W2   = (const float*)d_in[3];   // [768, 3072] f32
    float*       out  = (float*)d_out;           // [128, 768] f32

    float* partials = (float*)d_ws;                           // 128*8*768
    float* pooled   = partials + (size_t)NUM_SEQS * SPLITS * HIDDEN;  // 128*768
    float* h1       = pooled   + (size_t)NUM_SEQS * HIDDEN;           // 128*3072
    float* h2       = h1       + (size_t)NUM_SEQS * FF;               // 128*768

    // 1) ragged mean pool (bandwidth-bound, 201 MB stream)
    pool_partial_kernel<<<dim3(NUM_SEQS, SPLITS), 192, 0, stream>>>(hs, lens, partials);
    pool_reduce_kernel<<<NUM_SEQS, 192, 0, stream>>>(partials, lens, pooled);

    // 2) GEMM1: [128,768] x W1[3072,768]^T -> [128,3072]
    //    tiles = 8*192 = 1536, 8 waves/block -> 192 blocks
    wmma_gemm_nt_kernel<<<(8 * (FF / 16)) / 8, 256, 0, stream>>>(pooled, W1, h1, HIDDEN, FF);

    // 3) GEMM2: [128,3072] x W2[768,3072]^T -> [128,768]
    //    tiles = 8*48 = 384 -> 48 blocks
    wmma_gemm_nt_kernel<<<(8 * (HIDDEN / 16)) / 8, 256, 0, stream>>>(h1, W2, h2, FF, HIDDEN);

    // 4) L2 normalize rows -> d_out
    l2norm_rows_kernel<<<NUM_SEQS, 256, 0, stream>>>(h2, out);
}